// GATLayer_13692355740142
// MI455X (gfx1250) — compile-verified
//
#include <hip/hip_runtime.h>
#include <hip/hip_bf16.h>
#include <stdint.h>

// GAT layer: N=40000 nodes, E=640000 edges, IN=128, H=4 heads, C=32 (H*C=128)

typedef __attribute__((ext_vector_type(2))) float v2f;
typedef __attribute__((ext_vector_type(8))) float v8f;

#define IN_CH  128
#define OUT_CH 128
#define HEADS  4

// ---------------------------------------------------------------------------
// Helpers
// ---------------------------------------------------------------------------
__device__ __forceinline__ void atomicMaxF(float* addr, float val) {
  // classic signed/unsigned split: works for any mix once initialized to -inf
  if (val >= 0.0f) {
    atomicMax((int*)addr, __float_as_int(val));
  } else {
    atomicMin((unsigned int*)addr, __float_as_uint(val));
  }
}

__device__ __forceinline__ void atomicAddF(float* addr, float val) {
  // native global_atomic_add_f32 (no return, no CAS loop)
  unsafeAtomicAdd(addr, val);
}

// ---------------------------------------------------------------------------
// Kernel 1: xl = x @ W_lin + b_lin   via V_WMMA_F32_16X16X4_F32
//   block = 256 threads (8 waves); each wave computes a 16x128 output strip.
//   W_lin (128x128 = 64KB) staged in LDS once per block.
//   A loads are software-pipelined one K-step ahead to hide global latency.
// ---------------------------------------------------------------------------
__global__ __launch_bounds__(256) void gat_linear_wmma(
    const float* __restrict__ x, const float* __restrict__ W,
    const float* __restrict__ bias, float* __restrict__ xl, int n)
{
  __shared__ float Wl[IN_CH * OUT_CH];   // 64 KB

  const int tid = threadIdx.x;
  // cooperative load of W into LDS (float4, fully coalesced)
  for (int i = tid * 4; i < IN_CH * OUT_CH; i += 256 * 4) {
    *(float4*)&Wl[i] = *(const float4*)&W[i];
  }
  __syncthreads();

  const int wave  = tid >> 5;
  const int lane  = tid & 31;
  const int m16   = lane & 15;       // row within tile (A) / col within tile (B)
  const int khalf = lane >> 4;       // 0: K=0,1   1: K=2,3

  const long long base = ((long long)blockIdx.x * 8 + wave) * 16;
  if (base >= n) return;             // uniform per wave; after barrier

  v8f acc[8] = {};                   // 8 column tiles of 16x16 f32

  const float* xrow = x + (base + m16) * IN_CH + khalf * 2;

  // prefetch A fragment for kk=0
  float2 av = *(const float2*)(xrow);

  #pragma unroll 4
  for (int kk = 0; kk < IN_CH / 4; ++kk) {
    v2f a; a.x = av.x; a.y = av.y;
    // issue next A load before consuming WMMAs (depth-1 pipeline)
    if (kk + 1 < IN_CH / 4) av = *(const float2*)(xrow + (kk + 1) * 4);

    const int k0 = kk * 4 + khalf * 2;
    #pragma unroll
    for (int j = 0; j < 8; ++j) {
      const int ncol = j * 16 + m16;
      // B fragment: 4x16, lane holds W[k0][ncol], W[k0+1][ncol]
      v2f b;
      b.x = Wl[k0 * OUT_CH + ncol];
      b.y = Wl[(k0 + 1) * OUT_CH + ncol];
      acc[j] = __builtin_amdgcn_wmma_f32_16x16x4_f32(
          /*neg_a=*/false, a, /*neg_b=*/false, b,
          /*c_mod=*/(short)0, acc[j], /*reuse_a=*/false, /*reuse_b=*/false);
    }
  }

  // write back with bias.  C/D layout: vgpr r, lane l -> M = r + 8*(l>>4), N = l&15
  #pragma unroll
  for (int j = 0; j < 8; ++j) {
    const int ncol = j * 16 + m16;
    const float bv = bias[ncol];
    #pragma unroll
    for (int r = 0; r < 8; ++r) {
      const int mrow = r + khalf * 8;
      xl[(base + mrow) * OUT_CH + ncol] = acc[j][r] + bv;
    }
  }
}

// ---------------------------------------------------------------------------
// Kernel 2: per-node attention scores  s_i = xl @ W_att[:128], s_j = xl @ W_att[128:]
//   one wave per node; lane covers 4 feature dims; shuffle reduction.
// ---------------------------------------------------------------------------
__global__ __launch_bounds__(256) void gat_scores(
    const float* __restrict__ xl, const float* __restrict__ Watt,
    float* __restrict__ s_i, float* __restrict__ s_j, int n)
{
  const int wave = threadIdx.x >> 5;
  const int lane = threadIdx.x & 31;
  const long long node = (long long)blockIdx.x * 8 + wave;
  if (node >= n) return;

  const float4 xs = *(const float4*)(xl + node * OUT_CH + lane * 4);
  const float xv[4] = {xs.x, xs.y, xs.z, xs.w};

  float pi[HEADS] = {}, pj[HEADS] = {};
  #pragma unroll
  for (int t = 0; t < 4; ++t) {
    const int d = lane * 4 + t;
    const float4 wi = *(const float4*)(Watt + d * HEADS);
    const float4 wj = *(const float4*)(Watt + (OUT_CH + d) * HEADS);
    pi[0] += xv[t] * wi.x; pi[1] += xv[t] * wi.y;
    pi[2] += xv[t] * wi.z; pi[3] += xv[t] * wi.w;
    pj[0] += xv[t] * wj.x; pj[1] += xv[t] * wj.y;
    pj[2] += xv[t] * wj.z; pj[3] += xv[t] * wj.w;
  }
  #pragma unroll
  for (int off = 16; off > 0; off >>= 1) {
    #pragma unroll
    for (int h = 0; h < HEADS; ++h) {
      pi[h] += __shfl_xor(pi[h], off, 32);
      pj[h] += __shfl_xor(pj[h], off, 32);
    }
  }
  if (lane == 0) {
    *(float4*)(s_i + node * HEADS) = make_float4(pi[0], pi[1], pi[2], pi[3]);
    *(float4*)(s_j + node * HEADS) = make_float4(pj[0], pj[1], pj[2], pj[3]);
  }
}

// ---------------------------------------------------------------------------
// Kernel 3: init accumulators (out=0, amax=-inf, asum=0, deg=0)
// ---------------------------------------------------------------------------
__global__ void gat_init(float* __restrict__ out, float* __restrict__ amax,
                         float* __restrict__ asum, int* __restrict__ deg, int n)
{
  const long long i = (long long)blockIdx.x * 256 + threadIdx.x;
  if (i < (long long)n * OUT_CH) out[i] = 0.0f;
  if (i < (long long)n * HEADS) { amax[i] = -__builtin_inff(); asum[i] = 0.0f; }
  if (i < n) deg[i] = 0;
}

// ---------------------------------------------------------------------------
// Kernel 4: edge logits  alpha = leaky_relu(s_i[row]+s_j[col]+b_att), seg-max, deg
// ---------------------------------------------------------------------------
__global__ __launch_bounds__(256) void gat_edge_logits(
    const long long* __restrict__ ei, const float* __restrict__ s_i,
    const float* __restrict__ s_j, const float* __restrict__ b_att,
    float* __restrict__ alpha, float* __restrict__ amax, int* __restrict__ deg,
    int e)
{
  const int idx = blockIdx.x * 256 + threadIdx.x;
  if (idx >= e) return;
  const int r = (int)ei[idx];
  const int c = (int)ei[e + idx];

  const float4 si = *(const float4*)(s_i + (long long)r * HEADS);
  const float4 sj = *(const float4*)(s_j + (long long)c * HEADS);
  const float4 ba = *(const float4*)(b_att);

  float4 av;
  float* ap = (float*)&av;
  ap[0] = si.x + sj.x + ba.x;
  ap[1] = si.y + sj.y + ba.y;
  ap[2] = si.z + sj.z + ba.z;
  ap[3] = si.w + sj.w + ba.w;
  #pragma unroll
  for (int h = 0; h < HEADS; ++h) {
    ap[h] = (ap[h] >= 0.0f) ? ap[h] : 0.2f * ap[h];
    atomicMaxF(&amax[(long long)r * HEADS + h], ap[h]);
  }
  *(float4*)&alpha[(long long)idx * HEADS] = av;
  atomicAdd(&deg[c], 1);
}

// ---------------------------------------------------------------------------
// Kernel 5: w = exp(alpha - amax[row]); seg-sum
// ---------------------------------------------------------------------------
__global__ __launch_bounds__(256) void gat_edge_exp(
    const long long* __restrict__ ei, float* __restrict__ alpha,
    const float* __restrict__ amax, float* __restrict__ asum, int e)
{
  const int idx = blockIdx.x * 256 + threadIdx.x;
  if (idx >= e) return;
  const int r = (int)ei[idx];
  float4 av = *(float4*)&alpha[(long long)idx * HEADS];
  const float4 mx = *(const float4*)(amax + (long long)r * HEADS);
  float* ap = (float*)&av;
  const float* mp = (const float*)&mx;
  #pragma unroll
  for (int h = 0; h < HEADS; ++h) {
    const float w = expf(ap[h] - mp[h]);
    ap[h] = w;
    atomicAddF(&asum[(long long)r * HEADS + h], w);
  }
  *(float4*)&alpha[(long long)idx * HEADS] = av;
}

// ---------------------------------------------------------------------------
// Kernel 6: aggregate  out[col] += (w/asum[row]) * xl[row]   (wave per edge)
//   lane covers 4 channels -> head = lane>>3.  xl & out are L2-resident.
//   Edge index made wave-uniform via readfirstlane -> scalar s_load.
// ---------------------------------------------------------------------------
__global__ __launch_bounds__(256) void gat_aggregate(
    const long long* __restrict__ ei, const float* __restrict__ alpha,
    const float* __restrict__ asum, const float* __restrict__ xl,
    float* __restrict__ out, int e)
{
  const int wave = __builtin_amdgcn_readfirstlane(threadIdx.x >> 5);
  const int lane = threadIdx.x & 31;
  const long long eidx = (long long)blockIdx.x * 8 + wave;
  if (eidx >= e) return;
  const int r = (int)ei[eidx];        // wave-uniform -> scalar load
  const int c = (int)ei[e + eidx];    // wave-uniform -> scalar load
  const int h = lane >> 3;            // 32 lanes * 4 ch = 128 ch; head = (lane*4)/32

  const float coef = alpha[eidx * HEADS + h]
                   / (asum[(long long)r * HEADS + h] + 1e-16f);
  const float4 v = *(const float4*)(xl + (long long)r * OUT_CH + lane * 4);
  float* o = out + (long long)c * OUT_CH + lane * 4;
  atomicAddF(o + 0, coef * v.x);
  atomicAddF(o + 1, coef * v.y);
  atomicAddF(o + 2, coef * v.z);
  atomicAddF(o + 3, coef * v.w);
}

// ---------------------------------------------------------------------------
// Kernel 7: isolated nodes (deg==0) keep their own features
// ---------------------------------------------------------------------------
__global__ void gat_finalize(const int* __restrict__ deg,
                             const float* __restrict__ xl,
                             float* __restrict__ out, int n)
{
  const long long i = (long long)blockIdx.x * 256 + threadIdx.x;
  if (i >= (long long)n * OUT_CH) return;
  const int node = (int)(i >> 7);
  if (deg[node] == 0) out[i] = xl[i];
}

// ---------------------------------------------------------------------------
// Launch
// ---------------------------------------------------------------------------
extern "C" void kernel_launch(void* const* d_in, const int* in_sizes, int n_in,
                              void* d_out, int out_size, void* d_ws, size_t ws_size,
                              hipStream_t stream)
{
  const float*      x     = (const float*)d_in[0];
  const long long*  ei    = (const long long*)d_in[1];   // int64 [2, E]
  const float*      W_lin = (const float*)d_in[2];
  const float*      b_lin = (const float*)d_in[3];
  const float*      W_att = (const float*)d_in[4];
  const float*      b_att = (const float*)d_in[5];
  float*            out   = (float*)d_out;

  const int n = in_sizes[0] / IN_CH;   // 40000
  const int e = in_sizes[1] / 2;       // 640000

  // workspace layout (all 16B aligned by construction)
  char* ws = (char*)d_ws;
  float* xl    = (float*)ws;  ws += (size_t)n * OUT_CH * sizeof(float);
  float* s_i   = (float*)ws;  ws += (size_t)n * HEADS * sizeof(float);
  float* s_j   = (float*)ws;  ws += (size_t)n * HEADS * sizeof(float);
  float* amax  = (float*)ws;  ws += (size_t)n * HEADS * sizeof(float);
  float* asum  = (float*)ws;  ws += (size_t)n * HEADS * sizeof(float);
  float* alpha = (float*)ws;  ws += (size_t)e * HEADS * sizeof(float);
  int*   deg   = (int*)ws;    ws += (size_t)n * sizeof(int);

  const long long tot = (long long)n * OUT_CH;

  // 1) xl = x @ W_lin + b_lin  (WMMA f32: 16 rows/wave, 8 waves/block)
  gat_linear_wmma<<<(n + 127) / 128, 256, 0, stream>>>(x, W_lin, b_lin, xl, n);
  // 2) per-node scores
  gat_scores<<<(n + 7) / 8, 256, 0, stream>>>(xl, W_att, s_i, s_j, n);
  // 3) init accumulators (runs every call: atomics accumulate into d_out)
  gat_init<<<(int)((tot + 255) / 256), 256, 0, stream>>>(out, amax, asum, deg, n);
  // 4) edge logits + segment max + in-degree
  gat_edge_logits<<<(e + 255) / 256, 256, 0, stream>>>(ei, s_i, s_j, b_att,
                                                       alpha, amax, deg, e);
  // 5) exp + segment sum
  gat_edge_exp<<<(e + 255) / 256, 256, 0, stream>>>(ei, alpha, amax, asum, e);
  // 6) weighted scatter-aggregate (wave per edge)
  gat_aggregate<<<(e + 7) / 8, 256, 0, stream>>>(ei, alpha, asum, xl, out, e);
  // 7) isolated nodes copy their own features
  gat_finalize<<<(int)((tot + 255) / 256), 256, 0, stream>>>(deg, xl, out, n);
}